// EdgeEnhance_43954695307666
// MI455X (gfx1250) — compile-verified
//
#include <hip/hip_runtime.h>
#include <math.h>

// MI455X / gfx1250: wave32, WMMA f32 16x16x4 (fp32 precision matches reference).
// Convs = implicit GEMM, one wave: M=16 out-ch x N=64 pixels (4 accumulators),
// input row staged to LDS via global_load_async_to_lds_b32 (ASYNCcnt) and
// reused across KW taps + all 8 waves of the block.
//
// Fragment layouts per CDNA5 ISA 7.12.2 (f32 16x16x4):
//   A (16x4, 2 VGPR): lane l: row M=l&15; lanes<16 K={k0,k0+1}, lanes>=16 K={k0+2,k0+3}
//   B (4x16, 2 VGPR): lane l: col N=l&15; same K split
//   C/D (8 VGPR): VGPR r: lanes<16 -> M=r, lanes>=16 -> M=r+8; N=l&15
//
// Reference note: the combined Sobel kernel is identically zero, so
// edge = conv1x1(x, w_1x1).

typedef __attribute__((ext_vector_type(2))) float v2f;
typedef __attribute__((ext_vector_type(8))) float v8f;

#define HH 128
#define WW 128

__device__ __forceinline__ void async_copy_f32_to_lds(const float* lds_dst,
                                                      const float* gsrc) {
  unsigned int       ldsoff = (unsigned int)(unsigned long long)lds_dst; // low 32b = LDS offset
  unsigned long long gaddr  = (unsigned long long)gsrc;
  asm volatile("global_load_async_to_lds_b32 %0, %1, off"
               :: "v"(ldsoff), "v"(gaddr) : "memory");
}
__device__ __forceinline__ void wait_async_lds() {
  asm volatile("s_wait_asynccnt 0x0" ::: "memory");
}

// ---------------------------------------------------------------------------
// Implicit-GEMM conv + optional bias/BN/ReLU. Block = one (b,h) output row.
// blockDim = 32 * oc_tiles * 2 ; wave wv: ot = wv>>1, wt = wv&1 (64 px halves).
// STAGE: stage input row (IC*WW floats) in LDS per kh via async copies.
// out[((b*OC_tot + co_base + co)*HH + h)*WW + w]
// ---------------------------------------------------------------------------
template<int KH, int KW, int PAD, bool STAGE, bool HAS_BIAS, bool HAS_BN, bool RELU>
__global__ void conv_wmma_kernel(const float* __restrict__ in,
                                 const float* __restrict__ wgt,   // [OC][IC][KH][KW]
                                 const float* __restrict__ bias,
                                 const float* __restrict__ bn_s,
                                 const float* __restrict__ bn_b,
                                 float* __restrict__ out,
                                 int IC, int OC, int OC_tot, int co_base)
{
  extern __shared__ float lds[];        // [IC][WW] when STAGE
  const int tid  = threadIdx.x;
  const int lane = tid & 31;
  const int wv   = tid >> 5;
  const int ot   = wv >> 1;             // oc tile
  const int wt   = wv & 1;              // 64-pixel half of the row
  const int h    = blockIdx.x & (HH - 1);
  const int b    = blockIdx.x >> 7;     // HH == 128
  const int n    = lane & 15;
  const int half = lane >> 4;           // K pair {0,1} vs {2,3}
  const int co0  = ot * 16;
  const int wbase = wt * 64;
  const long HWs = (long)HH * WW;

  const float* inb = in + (long)b * IC * HWs;
  int co_m = co0 + n; if (co_m > OC - 1) co_m = OC - 1;   // clamp (masked on store)
  const float* wrow_base = wgt + (long)co_m * IC * (KH * KW);

  v8f acc[4];
#pragma unroll
  for (int j = 0; j < 4; ++j) acc[j] = (v8f){0.f,0.f,0.f,0.f,0.f,0.f,0.f,0.f};

  for (int kh = 0; kh < KH; ++kh) {
    const int hy = h + kh - PAD;
    const bool hok = (hy >= 0) && (hy < HH);    // block-uniform
    if (STAGE) {
      __syncthreads();                           // prior reads done before overwrite
      if (hok) {
        for (int e = tid; e < IC * WW; e += blockDim.x)
          async_copy_f32_to_lds(&lds[e],
              inb + (long)(e >> 7) * HWs + (long)hy * WW + (e & (WW - 1)));
        // warm next row in GL2 while async copies are in flight
        const int hy2 = hy + 1;
        if (kh + 1 < KH && hy2 < HH && hy2 >= 0) {
          for (int e = tid * 16; e < IC * WW; e += blockDim.x * 16)
            __builtin_prefetch(inb + (long)(e >> 7) * HWs + (long)hy2 * WW + (e & (WW - 1)), 0, 3);
        }
        wait_async_lds();                        // ASYNCcnt -> 0 (LDS written)
      }
      __syncthreads();                           // visible to all waves
    }
    if (!hok) continue;                          // zero row contributes nothing

    for (int kw = 0; kw < KW; ++kw) {
      const float* wrow = wrow_base + kh * KW + kw;
      for (int c0 = 0; c0 < IC; c0 += 4) {
        const int cA = c0 + half * 2;
        v2f a;
        a.x = wrow[(long)cA * (KH * KW)];
        a.y = wrow[(long)(cA + 1) * (KH * KW)];
#pragma unroll
        for (int j = 0; j < 4; ++j) {
          const int wx = wbase + j * 16 + n + kw - PAD;
          const bool wok = ((unsigned)wx < (unsigned)WW);
          const int wxc = wok ? wx : 0;
          v2f bf;
          if (STAGE) {
            bf.x = wok ? lds[cA * WW + wxc]       : 0.f;
            bf.y = wok ? lds[(cA + 1) * WW + wxc] : 0.f;
          } else {
            const float* src = inb + (long)hy * WW + wxc;
            bf.x = wok ? src[(long)cA * HWs]       : 0.f;
            bf.y = wok ? src[(long)(cA + 1) * HWs] : 0.f;
          }
          acc[j] = __builtin_amdgcn_wmma_f32_16x16x4_f32(
                       false, a, false, bf, (short)0, acc[j], false, false);
        }
      }
    }
  }

#pragma unroll
  for (int j = 0; j < 4; ++j) {
    const int wpix = wbase + j * 16 + n;
#pragma unroll
    for (int r = 0; r < 8; ++r) {
      const int co = co0 + r + half * 8;
      if (co < OC) {
        float v = acc[j][r];
        if (HAS_BIAS) v += bias[co];
        if (HAS_BN)   v = v * bn_s[co] + bn_b[co];
        if (RELU)     v = fmaxf(v, 0.f);
        out[(((long)b * OC_tot + co_base + co) * HH + h) * WW + wpix] = v;
      }
    }
  }
}

// ---------------------------------------------------------------------------
// Deformable 3x3 conv as implicit GEMM: K = tap(9) x channel(64).
// Bilinear corner indices/weights computed once per (tap,pixel), reused over c.
// ---------------------------------------------------------------------------
__global__ __launch_bounds__(256)
void deform_wmma_kernel(const float* __restrict__ fused,  // [B][64][H][W]
                        const float* __restrict__ offs,   // [B][18][H][W]
                        const float* __restrict__ w_def,  // [O][C][3][3]
                        const float* __restrict__ b_def,
                        float* __restrict__ df, int nb)
{
  const int C = 64, O = 64;
  const int lane = threadIdx.x & 31;
  const int wv   = threadIdx.x >> 5;
  const long total = (long)nb * HH * (WW / 16) * 4;
  long tile = (long)blockIdx.x * (blockDim.x >> 5) + wv;
  if (tile >= total) return;
  const int ot = (int)(tile % 4);
  long pt = tile / 4;
  const int wt = (int)(pt % (WW / 16));
  const int h  = (int)((pt / (WW / 16)) % HH);
  const int b  = (int)(pt / ((long)(WW / 16) * HH));

  const int n = lane & 15, half = lane >> 4, m = n;
  const int co0 = ot * 16;
  const int wpix = wt * 16 + n;
  const float* fb = fused + (long)b * C * HH * WW;

  v8f acc = {0.f,0.f,0.f,0.f,0.f,0.f,0.f,0.f};

  for (int t = 0; t < 9; ++t) {
    const int ky = t / 3 - 1, kx = t % 3 - 1;
    const long obase = (((long)b * 18 + 2 * t) * HH + h) * WW + wpix;
    const float py = offs[obase]                 + (float)(ky + h);
    const float px = offs[obase + (long)HH * WW] + (float)(kx + wpix);
    const float y0f = floorf(py), x0f = floorf(px);
    const float wy = py - y0f, wx = px - x0f;
    const int y0 = (int)y0f, x0 = (int)x0f, y1 = y0 + 1, x1 = x0 + 1;
    const bool vy0 = (y0 >= 0) && (y0 < HH), vy1 = (y1 >= 0) && (y1 < HH);
    const bool vx0 = (x0 >= 0) && (x0 < WW), vx1 = (x1 >= 0) && (x1 < WW);
    const int yc0 = min(max(y0, 0), HH - 1), yc1 = min(max(y1, 0), HH - 1);
    const int xc0 = min(max(x0, 0), WW - 1), xc1 = min(max(x1, 0), WW - 1);
    const int i00 = yc0 * WW + xc0, i01 = yc0 * WW + xc1;
    const int i10 = yc1 * WW + xc0, i11 = yc1 * WW + xc1;
    const float w00 = (vy0 && vx0) ? (1.f - wy) * (1.f - wx) : 0.f;
    const float w01 = (vy0 && vx1) ? (1.f - wy) * wx         : 0.f;
    const float w10 = (vy1 && vx0) ? wy * (1.f - wx)         : 0.f;
    const float w11 = (vy1 && vx1) ? wy * wx                 : 0.f;

    for (int c0 = 0; c0 < C; c0 += 4) {
      const int cA = c0 + half * 2;
      const float* p0 = fb + (long)cA * HH * WW;
      const float* p1 = p0 + HH * WW;
      v2f a, bf;
      bf.x = w00 * p0[i00] + w01 * p0[i01] + w10 * p0[i10] + w11 * p0[i11];
      bf.y = w00 * p1[i00] + w01 * p1[i01] + w10 * p1[i10] + w11 * p1[i11];
      a.x = w_def[((long)(co0 + m) * C + cA)     * 9 + t];
      a.y = w_def[((long)(co0 + m) * C + cA + 1) * 9 + t];
      acc = __builtin_amdgcn_wmma_f32_16x16x4_f32(
                false, a, false, bf, (short)0, acc, false, false);
    }
  }
#pragma unroll
  for (int r = 0; r < 8; ++r) {
    const int co = co0 + r + half * 8;
    df[(((long)b * O + co) * HH + h) * WW + wpix] = acc[r] + b_def[co];
  }
}

// ---------------------------------------------------------------------------
// Coordinate-attention small kernels (VALU; K is tiny)
// ---------------------------------------------------------------------------
__global__ void pool_kernel(const float* __restrict__ df, float* __restrict__ pool)
{
  int t = blockIdx.x * blockDim.x + threadIdx.x;      // 8*64*256
  if (t >= 8 * 64 * 256) return;
  int j = t & 255; int bc = t >> 8;                   // bc = b*64+c
  const float* base = df + (long)bc * HH * WW;
  float s = 0.f;
  if (j < HH) { const float* row = base + (long)j * WW;
                for (int w = 0; w < WW; ++w) s += row[w]; }
  else        { int w = j - HH;
                for (int hh = 0; hh < HH; ++hh) s += base[(long)hh * WW + w]; }
  pool[t] = s * (1.f / (float)WW);
}

__global__ void ca_mid_kernel(const float* __restrict__ pool,
                              const float* __restrict__ w1, const float* __restrict__ cb1,
                              const float* __restrict__ s1, const float* __restrict__ bb1,
                              float* __restrict__ y)
{
  int t = blockIdx.x * blockDim.x + threadIdx.x;      // 8*8*256
  if (t >= 8 * 8 * 256) return;
  int j = t & 255; int mm = (t >> 8) & 7; int b = t >> 11;
  float s = 0.f;
  for (int c = 0; c < 64; ++c) s += w1[mm * 64 + c] * pool[((long)b * 64 + c) * 256 + j];
  float v = (s + cb1[mm]) * s1[mm] + bb1[mm];
  y[((long)b * 8 + mm) * 256 + j] = fmaxf(v, 0.f);
}

__global__ void ca_attn_kernel(const float* __restrict__ y,
                               const float* __restrict__ wh, const float* __restrict__ bh,
                               const float* __restrict__ wwt, const float* __restrict__ bw,
                               float* __restrict__ ah, float* __restrict__ aw)
{
  int t = blockIdx.x * blockDim.x + threadIdx.x;      // 8*64*256
  if (t >= 8 * 64 * 256) return;
  int j = t & 255; int o = (t >> 8) & 63; int b = t >> 14;
  const float* wm = (j < HH) ? wh : wwt;
  float s = (j < HH) ? bh[o] : bw[o];
  for (int mm = 0; mm < 8; ++mm) s += wm[o * 8 + mm] * y[((long)b * 8 + mm) * 256 + j];
  float sig = 1.f / (1.f + __expf(-s));
  if (j < HH) ah[((long)b * 64 + o) * HH + j]        = sig;
  else        aw[((long)b * 64 + o) * WW + (j - HH)] = sig;
}

__global__ void apply_kernel(const float* __restrict__ df,
                             const float* __restrict__ ah, const float* __restrict__ aw,
                             const float* __restrict__ edge, float* __restrict__ tmp)
{
  long idx = (long)blockIdx.x * blockDim.x + threadIdx.x;   // 8*64*128*128
  if (idx >= (long)8 * 64 * HH * WW) return;
  int w = (int)(idx & (WW - 1));
  int h = (int)((idx >> 7) & (HH - 1));
  long bo = idx >> 14;                                      // b*64+o
  tmp[idx] = df[idx] * ah[bo * HH + h] * aw[bo * WW + w] + edge[idx];
}

// ---------------------------------------------------------------------------
// Launch
// ---------------------------------------------------------------------------
extern "C" void kernel_launch(void* const* d_in, const int* in_sizes, int n_in,
                              void* d_out, int out_size, void* d_ws, size_t ws_size,
                              hipStream_t stream) {
  const float* x      = (const float*)d_in[0];
  const float* w_1x1  = (const float*)d_in[1];
  const float* w3     = (const float*)d_in[2];
  const float* bn3_s  = (const float*)d_in[3];
  const float* bn3_b  = (const float*)d_in[4];
  const float* w5     = (const float*)d_in[5];
  const float* bn5_s  = (const float*)d_in[6];
  const float* bn5_b  = (const float*)d_in[7];
  const float* w7     = (const float*)d_in[8];
  const float* bn7_s  = (const float*)d_in[9];
  const float* bn7_b  = (const float*)d_in[10];
  const float* w_fuse = (const float*)d_in[11];
  const float* w_off  = (const float*)d_in[12];
  const float* b_off  = (const float*)d_in[13];
  const float* w_def  = (const float*)d_in[14];
  const float* b_def  = (const float*)d_in[15];
  const float* ca_w1  = (const float*)d_in[16];
  const float* ca_cb1 = (const float*)d_in[17];
  const float* ca_s1  = (const float*)d_in[18];
  const float* ca_bb1 = (const float*)d_in[19];
  const float* ca_wh  = (const float*)d_in[20];
  const float* ca_bh  = (const float*)d_in[21];
  const float* ca_ww  = (const float*)d_in[22];
  const float* ca_bw  = (const float*)d_in[23];
  const float* w_out  = (const float*)d_in[24];
  const float* b_out  = (const float*)d_in[25];
  const float* bno_s  = (const float*)d_in[26];
  const float* bno_b  = (const float*)d_in[27];
  float* ws  = (float*)d_ws;
  float* out = (float*)d_out;

  // workspace layout (floats); tmp reuses the branch-concat buffer
  const size_t WS_A     = 0;                         // [8][192][128][128]
  const size_t WS_FUSED = WS_A     + (size_t)8 * 192 * HH * WW;
  const size_t WS_EDGE  = WS_FUSED + (size_t)8 * 64  * HH * WW;
  const size_t WS_OFF   = WS_EDGE  + (size_t)8 * 64  * HH * WW;
  const size_t WS_DF    = WS_OFF   + (size_t)8 * 18  * HH * WW;
  const size_t WS_POOL  = WS_DF    + (size_t)8 * 64  * HH * WW;
  const size_t WS_Y     = WS_POOL  + (size_t)8 * 64 * 256;
  const size_t WS_AH    = WS_Y     + (size_t)8 * 8 * 256;
  const size_t WS_AW    = WS_AH    + (size_t)8 * 64 * HH;
  const size_t WS_TMP   = WS_A;

  const int rows   = 8 * HH;                 // one block per output row
  const int shmem64 = 64 * WW * sizeof(float);   // 32 KB staged row (IC=64)

  // branch convs (BN+ReLU) -> concatenated [B,192,H,W] buffer
  conv_wmma_kernel<3,3,1,true,false,true,true><<<rows, 256, shmem64, stream>>>(
      x, w3, nullptr, bn3_s, bn3_b, ws + WS_A, 64, 64, 192, 0);
  conv_wmma_kernel<5,5,2,true,false,true,true><<<rows, 256, shmem64, stream>>>(
      x, w5, nullptr, bn5_s, bn5_b, ws + WS_A, 64, 64, 192, 64);
  conv_wmma_kernel<7,7,3,true,false,true,true><<<rows, 256, shmem64, stream>>>(
      x, w7, nullptr, bn7_s, bn7_b, ws + WS_A, 64, 64, 192, 128);
  // edge = conv1x1(x, w_1x1) (Sobel term is identically zero); no staging for 1x1
  conv_wmma_kernel<1,1,0,false,false,false,false><<<rows, 256, 0, stream>>>(
      x, w_1x1, nullptr, nullptr, nullptr, ws + WS_EDGE, 64, 64, 64, 0);
  // fuse: 1x1 GEMM over 192 channels
  conv_wmma_kernel<1,1,0,false,false,false,false><<<rows, 256, 0, stream>>>(
      ws + WS_A, w_fuse, nullptr, nullptr, nullptr, ws + WS_FUSED, 192, 64, 64, 0);
  // offset conv: 64 -> 18 + bias (2 oc tiles -> 4 waves/block)
  conv_wmma_kernel<3,3,1,true,true,false,false><<<rows, 128, shmem64, stream>>>(
      ws + WS_FUSED, w_off, b_off, nullptr, nullptr, ws + WS_OFF, 64, 18, 18, 0);
  // deformable conv
  const int tiles64 = 8 * HH * (WW / 16) * 4;
  deform_wmma_kernel<<<tiles64 / 8, 256, 0, stream>>>(
      ws + WS_FUSED, ws + WS_OFF, w_def, b_def, ws + WS_DF, 8);
  // coordinate attention
  pool_kernel<<<(8 * 64 * 256) / 256, 256, 0, stream>>>(ws + WS_DF, ws + WS_POOL);
  ca_mid_kernel<<<(8 * 8 * 256) / 256, 256, 0, stream>>>(
      ws + WS_POOL, ca_w1, ca_cb1, ca_s1, ca_bb1, ws + WS_Y);
  ca_attn_kernel<<<(8 * 64 * 256) / 256, 256, 0, stream>>>(
      ws + WS_Y, ca_wh, ca_bh, ca_ww, ca_bw, ws + WS_AH, ws + WS_AW);
  // out = df * a_h * a_w + edge   (into tmp, reusing branch buffer)
  apply_kernel<<<(int)(((long)8 * 64 * HH * WW) / 256), 256, 0, stream>>>(
      ws + WS_DF, ws + WS_AH, ws + WS_AW, ws + WS_EDGE, ws + WS_TMP);
  // final conv3x3 + bias + BN + ReLU -> d_out
  conv_wmma_kernel<3,3,1,true,true,true,true><<<rows, 256, shmem64, stream>>>(
      ws + WS_TMP, w_out, b_out, bno_s, bno_b, out, 64, 64, 64, 0);
}